// HierarchicalNeuralFeedback_87935160418556
// MI455X (gfx1250) — compile-verified
//
#include <hip/hip_runtime.h>
#include <hip/hip_bf16.h>
#include <math.h>
#include <stdint.h>

typedef __attribute__((ext_vector_type(2))) float v2f;
typedef __attribute__((ext_vector_type(4))) float v4f;
typedef __attribute__((ext_vector_type(8))) float v8f;

#define TILE_ROWS 64
#define TILE_K    64
#define LDS_STRIDE 68   // 64 + 4 pad: conflict-free b64 fragment reads

// ---------------------------------------------------------------------------
// Layer 1: l1 = 0.95*l1s + 0.05*tanh(sm)   (8192 elements)
// ---------------------------------------------------------------------------
__global__ void l1_kernel(const float* __restrict__ sm,
                          const float* __restrict__ l1s,
                          float* __restrict__ out_l1) {
    int i = blockIdx.x * blockDim.x + threadIdx.x;
    out_l1[i] = 0.95f * l1s[i] + 0.05f * tanhf(sm[i]);
}

// ---------------------------------------------------------------------------
// Split-K GEMV via V_WMMA_F32_16X16X4_F32 with double-buffered
// GLOBAL_LOAD_ASYNC_TO_LDS_B128 tile staging (ASYNCcnt pipelined).
//
//   grid.x = M/64 row blocks, grid.y = SPLITK, block = 128 threads (4 waves).
//   Each wave owns 16 rows. Per tile, each wave issues exactly 8 async-to-LDS
//   b128 instructions, so "s_wait_asynccnt 8" = previous tile complete while
//   the next tile streams. A fragment (16x4 f32): lanes 0-15 -> cols {k,k+1},
//   lanes 16-31 -> cols {k+2,k+3}. B fragment: column 0 only (lane0/lane16).
//   D column 0: lane0 c[0..7] = rows base..+7, lane16 c[0..7] = base+8..+15.
// ---------------------------------------------------------------------------
__launch_bounds__(128)
__global__ void gemv_wmma_kernel(const float* __restrict__ W,
                                 const float* __restrict__ x,
                                 float* __restrict__ part,
                                 int M, int K, int ksliceLen) {
    __shared__ float tile[2][TILE_ROWS * LDS_STRIDE];
    __shared__ float xs[2][TILE_K];

    const int tid  = threadIdx.x;
    const int lane = tid & 31;
    const int wave = tid >> 5;                 // 0..3
    const int half = lane >> 4;                // 0/1: K-pair select
    const int r16  = lane & 15;
    const int rowBase = blockIdx.x * TILE_ROWS;
    const int k0      = blockIdx.y * ksliceLen;
    const int ldsRow  = wave * 16 + r16;
    const int nTiles  = ksliceLen / TILE_K;

    // Issue one 64x64 f32 tile as 8 async b128 loads per wave.
    // 16 consecutive threads cover one 256B row segment -> full cachelines.
    auto issue_tile = [&](int buf, int kt) {
        const float* base = W + (size_t)rowBase * K + (size_t)(k0 + kt);
        unsigned ldsBase = (unsigned)(uintptr_t)&tile[buf][0];
        #pragma unroll
        for (int i = 0; i < 8; ++i) {
            int idx = (tid + i * 128) * 4;     // 0..4092
            int r = idx >> 6;
            int c = idx & 63;
            unsigned long long src =
                (unsigned long long)(uintptr_t)(base + (size_t)r * K + c);
            unsigned dst = ldsBase + (unsigned)((r * LDS_STRIDE + c) * 4);
            asm volatile("global_load_async_to_lds_b128 %0, %1, off"
                         :: "v"(dst), "v"(src)
                         : "memory");
        }
    };

    v8f acc = {0.f, 0.f, 0.f, 0.f, 0.f, 0.f, 0.f, 0.f};

    issue_tile(0, 0);                          // prologue: tile 0 in flight

    for (int t = 0; t < nTiles; ++t) {
        const int cur = t & 1;

        if (t + 1 < nTiles) {
            issue_tile(cur ^ 1, (t + 1) * TILE_K);      // stream next tile
            if (tid < TILE_K) xs[cur][tid] = x[k0 + t * TILE_K + tid];
            asm volatile("s_wait_asynccnt 0x8" ::: "memory");  // tile t done
        } else {
            if (tid < TILE_K) xs[cur][tid] = x[k0 + t * TILE_K + tid];
            asm volatile("s_wait_asynccnt 0x0" ::: "memory");  // drain
        }
        __syncthreads();                       // cross-wave LDS visibility

        const float* tb = &tile[cur][0];
        const float* xb = &xs[cur][0];
        #pragma unroll
        for (int kk = 0; kk < TILE_K; kk += 4) {
            v2f a = *(const v2f*)&tb[ldsRow * LDS_STRIDE + kk + 2 * half];
            v2f b = {0.f, 0.f};
            if (r16 == 0) {                    // lanes 0 and 16 carry x
                b.x = xb[kk + 2 * half];
                b.y = xb[kk + 2 * half + 1];
            }
            acc = __builtin_amdgcn_wmma_f32_16x16x4_f32(
                false, a, false, b, (short)0, acc, false, false);
        }
        __syncthreads();                       // safe to overwrite buffer
    }

    if (r16 == 0) {                            // lanes 0,16: column 0 of D
        int row = rowBase + wave * 16 + half * 8;
        float* pp = part + (size_t)blockIdx.y * M + row;
        #pragma unroll
        for (int i = 0; i < 8; ++i) pp[i] = acc[i];
    }
}

// ---------------------------------------------------------------------------
// Epilogues: sum split-K partials, tanh-decay. 0.3 top-down factor folded here.
// ---------------------------------------------------------------------------
__global__ void l2_epilogue(const float* __restrict__ part12,   // [8][4096]
                            const float* __restrict__ part32,   // [4][4096]
                            const float* __restrict__ l2s,
                            float* __restrict__ out_l2) {
    int m = blockIdx.x * blockDim.x + threadIdx.x;               // < 4096
    float s12 = 0.f, s32 = 0.f;
    #pragma unroll
    for (int i = 0; i < 8; ++i) s12 += part12[i * 4096 + m];
    #pragma unroll
    for (int i = 0; i < 4; ++i) s32 += part32[i * 4096 + m];
    float s = s12 + 0.3f * s32;
    out_l2[m] = 0.98f * l2s[m] + 0.02f * tanhf(s);
}

__global__ void l3_epilogue(const float* __restrict__ part23,   // [8][2048]
                            const float* __restrict__ l3s,
                            float* __restrict__ out_l3) {
    int m = blockIdx.x * blockDim.x + threadIdx.x;               // < 2048
    float s = 0.f;
    #pragma unroll
    for (int i = 0; i < 8; ++i) s += part23[i * 2048 + m];
    out_l3[m] = 0.99f * l3s[m] + 0.01f * tanhf(s);
}

// ---------------------------------------------------------------------------
// Scalars: correlations, coherence, hierarchy activity. One 1024-thread block.
// ---------------------------------------------------------------------------
__launch_bounds__(1024)
__global__ void scalars_kernel(const float* __restrict__ out,
                               const float* __restrict__ corr_in,
                               const float* __restrict__ coh_in,
                               float* __restrict__ dst) {
    const float* l1 = out;
    const float* l2 = out + 8192;
    const float* l3 = out + 12288;

    float acc[13];
    #pragma unroll
    for (int j = 0; j < 13; ++j) acc[j] = 0.f;

    for (int i = threadIdx.x; i < 14336; i += 1024) {
        float v = (i < 8192) ? l1[i] : ((i < 12288) ? l2[i - 8192] : l3[i - 12288]);
        acc[0] += fabsf(v);
        acc[1] += v;
        acc[2] += v * v;
        if (i < 4096) {
            float a = l1[i], b = l2[i];
            acc[3] += a; acc[4] += b; acc[5] += a * a; acc[6] += b * b; acc[7] += a * b;
        }
        if (i < 2048) {
            float a = l2[i], b = l3[i];
            acc[8] += a; acc[9] += b; acc[10] += a * a; acc[11] += b * b; acc[12] += a * b;
        }
    }

    // wave32 reduction, then cross-wave via LDS
    #pragma unroll
    for (int off = 16; off > 0; off >>= 1)
        #pragma unroll
        for (int j = 0; j < 13; ++j)
            acc[j] += __shfl_xor(acc[j], off, 32);

    __shared__ float red[32][13];
    int lane = threadIdx.x & 31, wv = threadIdx.x >> 5;
    if (lane == 0)
        for (int j = 0; j < 13; ++j) red[wv][j] = acc[j];
    __syncthreads();

    if (threadIdx.x == 0) {
        float t[13];
        #pragma unroll
        for (int j = 0; j < 13; ++j) t[j] = 0.f;
        for (int w = 0; w < 32; ++w)
            for (int j = 0; j < 13; ++j) t[j] += red[w][j];

        const float n12 = 4096.f, n23 = 2048.f, ntot = 14336.f;
        float num12 = t[7] - t[3] * t[4] / n12;
        float den12 = sqrtf((t[5] - t[3] * t[3] / n12) * (t[6] - t[4] * t[4] / n12));
        float c12 = num12 / den12;
        if (c12 != c12) c12 = 0.f;                       // NaN -> 0
        float num23 = t[12] - t[8] * t[9] / n23;
        float den23 = sqrtf((t[10] - t[8] * t[8] / n23) * (t[11] - t[9] * t[9] / n23));
        float c23 = num23 / den23;
        if (c23 != c23) c23 = 0.f;

        float corr_new = 0.9f * corr_in[0] + 0.05f * (c12 + c23);

        float total = t[0];
        float var = (t[2] - t[1] * t[1] / ntot) / (ntot - 1.f);  // ddof=1
        float coh = total / (var + 1e-6f);
        coh = fminf(fmaxf(coh, 0.f), 10.f);
        float coh_new = 0.9f * coh_in[0] + 0.1f * coh;

        dst[14336] = corr_new;
        dst[14337] = coh_new;
        dst[14338] = total / ntot;
    }
}

// ---------------------------------------------------------------------------
// Launch: l1 -> [w12@l1, w32@l3s] -> l2 -> w23@l2 -> l3 -> scalars
// ---------------------------------------------------------------------------
extern "C" void kernel_launch(void* const* d_in, const int* in_sizes, int n_in,
                              void* d_out, int out_size, void* d_ws, size_t ws_size,
                              hipStream_t stream) {
    const float* sm   = (const float*)d_in[0];   // 8192
    const float* l1s  = (const float*)d_in[1];   // 8192
    const float* l2s  = (const float*)d_in[2];   // 4096
    const float* l3s  = (const float*)d_in[3];   // 2048
    const float* w12  = (const float*)d_in[4];   // 4096 x 8192
    const float* w23  = (const float*)d_in[5];   // 2048 x 4096
    const float* w32  = (const float*)d_in[6];   // 4096 x 2048
    const float* corr = (const float*)d_in[8];
    const float* coh  = (const float*)d_in[9];

    float* out    = (float*)d_out;
    float* out_l1 = out;                // 8192
    float* out_l2 = out + 8192;         // 4096
    float* out_l3 = out + 12288;        // 2048

    float* ws     = (float*)d_ws;
    float* part12 = ws;                 // [8][4096]
    float* part32 = ws + 32768;         // [4][4096]
    float* part23 = ws + 49152;         // [8][2048]

    // Layer 1
    l1_kernel<<<dim3(32), dim3(256), 0, stream>>>(sm, l1s, out_l1);

    // w12 @ l1 : M=4096, K=8192, 8-way split-K (slice 1024) -> 512 WGs
    gemv_wmma_kernel<<<dim3(64, 8), dim3(128), 0, stream>>>(
        w12, out_l1, part12, 4096, 8192, 1024);

    // w32 @ l3s : M=4096, K=2048, 4-way split (slice 512) -> 256 WGs
    gemv_wmma_kernel<<<dim3(64, 4), dim3(128), 0, stream>>>(
        w32, l3s, part32, 4096, 2048, 512);

    // Layer 2 (0.3 feedback factor applied here)
    l2_epilogue<<<dim3(16), dim3(256), 0, stream>>>(part12, part32, l2s, out_l2);

    // w23 @ l2 : M=2048, K=4096, 8-way split (slice 512) -> 256 WGs
    gemv_wmma_kernel<<<dim3(32, 8), dim3(128), 0, stream>>>(
        w23, out_l2, part23, 2048, 4096, 512);

    // Layer 3
    l3_epilogue<<<dim3(8), dim3(256), 0, stream>>>(part23, l3s, out_l3);

    // Scalars
    scalars_kernel<<<dim3(1), dim3(1024), 0, stream>>>(out, corr, coh, out);
}